// GCN_79723182948735
// MI455X (gfx1250) — compile-verified
//
#include <hip/hip_runtime.h>
#include <hip/hip_bf16.h>

#if defined(__HIP_DEVICE_COMPILE__) && !__has_builtin(__builtin_amdgcn_wmma_f32_16x16x4_f32)
#error "missing __builtin_amdgcn_wmma_f32_16x16x4_f32 on device toolchain"
#endif

typedef __attribute__((ext_vector_type(2))) float v2f;
typedef __attribute__((ext_vector_type(8))) float v8f;

#define FEAT_IN  65
#define FEAT_HID 64
#define FEAT_OUT 64
#define KPAD     68      // FEAT_IN rounded up to multiple of 4

// ---------------------------------------------------------------------------
// Kernel 1: seed accumulator h0 = x  (folds the "+ x" of GIN eps=0)
// ---------------------------------------------------------------------------
__global__ __launch_bounds__(256) void gin_init_kernel(const float* __restrict__ x,
                                                       float* __restrict__ h0,
                                                       int total) {
    int i = blockIdx.x * blockDim.x + threadIdx.x;
    int nv = total >> 2;
    if (i < nv) {
        reinterpret_cast<float4*>(h0)[i] = reinterpret_cast<const float4*>(x)[i];
    }
    int rem = total & 3;
    if (i < rem) {
        int base = nv << 2;
        h0[base + i] = x[base + i];
    }
}

// ---------------------------------------------------------------------------
// Kernel 2: scatter-sum  h0[dst] += x[src]   (one wave32 per edge)
// ---------------------------------------------------------------------------
__global__ __launch_bounds__(256) void gin_scatter_kernel(const float* __restrict__ x,
                                                          const int* __restrict__ ei,
                                                          float* __restrict__ h0,
                                                          int E) {
    int gw   = (blockIdx.x * 256 + threadIdx.x) >> 5;  // wave id == edge id
    int lane = threadIdx.x & 31;
    if (gw >= E) return;
    int src = ei[gw];       // row 0 of edge_index
    int dst = ei[E + gw];   // row 1 of edge_index
    const float* xs = x  + (size_t)src * FEAT_IN;
    float*       hd = h0 + (size_t)dst * FEAT_IN;
    for (int f = lane; f < FEAT_IN; f += 32) {
        unsafeAtomicAdd(&hd[f], xs[f]);  // global_atomic_add_f32
    }
}

// ---------------------------------------------------------------------------
// Kernel 3: fused MLP  out = relu(h0 @ W1 + b1) @ W2 + b2  via WMMA f32 16x16x4
// 128 threads = 4 waves; each wave owns a 16-row tile, full N=64 in 4 accs.
// Weights packed in LDS as K-pair-interleaved so B frags are one ds_load_b64.
// ---------------------------------------------------------------------------
__global__ __launch_bounds__(128) void gin_mlp_kernel(const float* __restrict__ h0,
                                                      const float* __restrict__ W1,
                                                      const float* __restrict__ b1,
                                                      const float* __restrict__ W2,
                                                      const float* __restrict__ b2,
                                                      float* __restrict__ out,
                                                      int n) {
    // sW1p[(k/2)*128 + 2*c + (k&1)] = W1[k][c], K padded to 68 with zeros
    __shared__ float sW1p[(KPAD / 2) * 2 * FEAT_HID];     // 34*128 = 17408 B
    __shared__ float sW2p[(FEAT_HID / 2) * 2 * FEAT_OUT]; // 32*128 = 16384 B
    __shared__ float sBuf[4][16 * KPAD];                  // per-wave tile, 4*4352 B

    const int tid = threadIdx.x;
    for (int i = tid; i < KPAD * FEAT_HID; i += 128) {
        int k = i >> 6, c = i & 63;
        float v = (k < FEAT_IN) ? W1[k * FEAT_HID + c] : 0.f;
        sW1p[(k >> 1) * 128 + (c << 1) + (k & 1)] = v;
    }
    for (int i = tid; i < FEAT_HID * FEAT_OUT; i += 128) {
        int k = i >> 6, c = i & 63;
        sW2p[(k >> 1) * 128 + (c << 1) + (k & 1)] = W2[k * FEAT_OUT + c];
    }
    __syncthreads();

    const int wave = tid >> 5;
    const int lane = tid & 31;
    const int half = lane >> 4;   // 0 or 1
    const int l16  = lane & 15;
    const int tileRow = (blockIdx.x * 4 + wave) * 16;
    float* myB = sBuf[wave];

    // ---- stage 16x68 input tile, coalesced, branch-free (clamp + select) ----
    for (int i = lane; i < 16 * KPAD; i += 32) {
        int r = i / KPAD, c = i - r * KPAD;
        int gr = tileRow + r;
        int rc = gr < n ? gr : (n - 1);
        int cc = c < FEAT_IN ? c : (FEAT_IN - 1);
        float v = h0[rc * FEAT_IN + cc];
        myB[i] = (gr < n && c < FEAT_IN) ? v : 0.f;
    }
    // intra-wave LDS ops are processed in order; no cross-wave sharing of myB

    // ---------------- layer 1: [16x68] @ [68x64] ---------------------------
    v8f acc[4] = {};
    for (int s = 0; s < KPAD / 4; ++s) {
        const int kk = s * 4 + half * 2;             // always even
        v2f a = *(const v2f*)&myB[l16 * KPAD + kk];  // ds_load_b64
        const float* w = &sW1p[(kk >> 1) * 128];
#pragma unroll
        for (int j = 0; j < 4; ++j) {
            v2f b = *(const v2f*)&w[(j * 16 + l16) * 2];  // ds_load_b64
            acc[j] = __builtin_amdgcn_wmma_f32_16x16x4_f32(
                false, a, false, b, (short)0, acc[j], false, false);
        }
    }

    // bias + ReLU -> overwrite tile buffer (cols 0..63; layer-2 K never reads 64+)
#pragma unroll
    for (int j = 0; j < 4; ++j) {
        const int col = j * 16 + l16;
        const float bb = b1[col];
#pragma unroll
        for (int v = 0; v < 8; ++v) {
            const int m = v + half * 8;
            float hv = acc[j][v] + bb;
            myB[m * KPAD + col] = hv > 0.f ? hv : 0.f;
        }
    }

    // ---------------- layer 2: [16x64] @ [64x64] ---------------------------
    v8f acc2[4] = {};
    for (int s = 0; s < FEAT_HID / 4; ++s) {
        const int kk = s * 4 + half * 2;
        v2f a = *(const v2f*)&myB[l16 * KPAD + kk];
        const float* w = &sW2p[(kk >> 1) * 128];
#pragma unroll
        for (int j = 0; j < 4; ++j) {
            v2f b = *(const v2f*)&w[(j * 16 + l16) * 2];
            acc2[j] = __builtin_amdgcn_wmma_f32_16x16x4_f32(
                false, a, false, b, (short)0, acc2[j], false, false);
        }
    }

    // bias + store
#pragma unroll
    for (int j = 0; j < 4; ++j) {
        const int col = j * 16 + l16;
        const float bb = b2[col];
#pragma unroll
        for (int v = 0; v < 8; ++v) {
            const int r = tileRow + v + half * 8;
            if (r < n) out[r * FEAT_OUT + col] = acc2[j][v] + bb;
        }
    }
}

// ---------------------------------------------------------------------------
extern "C" void kernel_launch(void* const* d_in, const int* in_sizes, int n_in,
                              void* d_out, int out_size, void* d_ws, size_t ws_size,
                              hipStream_t stream) {
    const float* x  = (const float*)d_in[0];
    const int*   ei = (const int*)  d_in[1];   // [2, E] (JAX default int32)
    const float* W1 = (const float*)d_in[2];   // [65,64] row-major (K x N)
    const float* b1 = (const float*)d_in[3];
    const float* W2 = (const float*)d_in[4];   // [64,64]
    const float* b2 = (const float*)d_in[5];
    float* out = (float*)d_out;

    const int n = in_sizes[0] / FEAT_IN;   // 100000 nodes
    const int E = in_sizes[1] / 2;         // 1600000 edges
    float* h0 = (float*)d_ws;              // [n, 65] accumulator

    // 1) h0 = x
    {
        int total = n * FEAT_IN;
        int work  = (total >> 2) + 4;  // vec4 body + tail
        int blocks = (work + 255) / 256;
        gin_init_kernel<<<blocks, 256, 0, stream>>>(x, h0, total);
    }
    // 2) h0[dst] += x[src]  (one wave per edge)
    {
        long long threads = (long long)E * 32;
        int blocks = (int)((threads + 255) / 256);
        gin_scatter_kernel<<<blocks, 256, 0, stream>>>(x, ei, h0, E);
    }
    // 3) out = relu(h0 @ W1 + b1) @ W2 + b2
    {
        int blocks = (n + 63) / 64;    // 64 rows / block (4 waves x 16)
        gin_mlp_kernel<<<blocks, 128, 0, stream>>>(h0, W1, b1, W2, b2, out, n);
    }
}